// Capsule3D_4741643895374
// MI455X (gfx1250) — compile-verified
//
#include <hip/hip_runtime.h>
#include <hip/hip_bf16.h>
#include <math.h>

typedef __attribute__((ext_vector_type(2))) float v2f;
typedef __attribute__((ext_vector_type(8))) float v8f;
typedef unsigned int u32x4 __attribute__((ext_vector_type(4)));
typedef int i32x4 __attribute__((ext_vector_type(4)));
typedef int i32x8 __attribute__((ext_vector_type(8)));

#define B_   16
#define H_   32
#define W_   32
#define IC_  32
#define IL_  8
#define OH_  30
#define OW_  30
#define C_   16
#define L_   8
#define NCH  128            // C_*L_
#define KTOT 72             // 3*3*IL_
#define P_   (OH_*OW_)      // 900
#define BI_  (B_*IC_)       // 512
#define MTOT (BI_*P_)       // 460800
#define J_   (P_*C_)        // 14400
#define ASTRIDE 76          // padded K-stride for A tiles (16B aligned, conflict-free)

// ------------------------------------------------------------------
// Kernel 1: implicit-GEMM 3x3 VALID conv via V_WMMA_F32_16X16X4_F32.
//   u_hat[bi][p][n] = sum_{ky,kx,l} x[b, oy+ky, ox+kx, i, l] * w[l,ky,kx,0,n]
//   M = 460800 (bi*900+p), N = 128, K = 72.
// Block = 128 threads (4 waves). Each wave owns one 16-row M tile and
// sweeps all 8 N-tiles. Weights staged once per block via the Tensor
// Data Mover (TENSOR_LOAD_TO_LDS) as a 3D tile copy that performs the
// (l,tap) -> k=tap*8+l reorder in-flight.
// ------------------------------------------------------------------
__global__ __launch_bounds__(128)
void conv_wmma_kernel(const float* __restrict__ x, const float* __restrict__ w,
                      float* __restrict__ uhat) {
  __shared__ float Bs[KTOT][NCH];          // 36 KB  B[k][n]
  __shared__ float As[4][16][ASTRIDE];     // 19 KB  A[tile][m][k]

  const int tid  = threadIdx.x;
  const int wave = tid >> 5;
  const int lane = tid & 31;
  const int half = lane >> 4;
  const int lm   = lane & 15;

#if __has_builtin(__builtin_amdgcn_tensor_load_to_lds)
  // --- TDM staging of weights ---------------------------------------
  // Source tensor view of w (floats): addr(n,l,tap) = n + l*1152 + tap*128
  // TDM walks X=dim0 (n,128) fastest, then Y=dim1 (l,8, stride 1152),
  // then Z=dim2 (tap,9, stride 128), writing LDS contiguously ->
  // LDS order = ((tap*8)+l)*128 + n = Bs[k][n] with k = tap*8+l.  (ISA ch.8)
  if (wave == 0) {
    const unsigned long long gaddr = (unsigned long long)(uintptr_t)w;
    const unsigned int lds_off = (unsigned int)(uintptr_t)(&Bs[0][0]); // low 32 bits = LDS byte offset

    u32x4 g0;
    g0.x = 1u;                                        // count=1, is_restore=0, gather off
    g0.y = lds_off;                                   // lds_addr
    g0.z = (unsigned int)gaddr;                       // global_addr[31:0]
    g0.w = (unsigned int)(gaddr >> 32) | (2u << 30);  // global_addr[56:32] | type=2

    i32x8 g1;
    g1[0] = 0x00020000;          // wg_mask=0 | data_size=2 (4B) | no flags
    g1[1] = (int)(128u << 16);   // tensor_dim0[15:0] -> bits[63:48]  (=128)
    g1[2] = (int)(8u   << 16);   // tensor_dim0 hi=0 | tensor_dim1 lo=8
    g1[3] = (int)(128u << 16);   // tensor_dim1 hi=0 | tile_dim0=128
    g1[4] = (int)(8u | (9u << 16)); // tile_dim1=8 | tile_dim2=9
    g1[5] = 1152;                // tensor_dim0_stride[31:0]
    g1[6] = (int)(128u << 16);   // stride0 hi=0 | tensor_dim1_stride lo=128
    g1[7] = 0;                   // tensor_dim1_stride hi=0

    i32x4 g2; g2[0] = 9; g2[1] = 0; g2[2] = 0; g2[3] = 0;  // tensor_dim2=9 (3D)
    i32x4 g3; g3[0] = 0; g3[1] = 0; g3[2] = 0; g3[3] = 0;

#if defined(__clang_major__) && (__clang_major__ >= 23)
    i32x8 gpad = {0, 0, 0, 0, 0, 0, 0, 0};
    __builtin_amdgcn_tensor_load_to_lds(g0, g1, g2, g3, gpad, 0);
#else
    __builtin_amdgcn_tensor_load_to_lds(g0, g1, g2, g3, 0);
#endif
    __builtin_amdgcn_s_wait_tensorcnt(0);
  }
#else
  // --- fallback: loop staging of weights ----------------------------
  for (int idx = tid; idx < KTOT * NCH; idx += 128) {
    int k   = idx >> 7;
    int n   = idx & 127;
    int tap = k >> 3;
    int l   = k & 7;
    Bs[k][n] = w[(l * 9 + tap) * NCH + n];
  }
#endif

  // Stage A: 4 tiles x 16 rows x 9 taps; each tap = 8 contiguous floats of x.
  const int m0 = blockIdx.x * 64;
  for (int u = tid; u < 4 * 16 * 9; u += 128) {
    int tile = u / 144;
    int rr   = (u / 9) & 15;
    int tap  = u % 9;
    int m  = m0 + tile * 16 + rr;
    int bi = m / P_;
    int p  = m - bi * P_;
    int b  = bi >> 5;
    int i  = bi & 31;
    int oy = p / OW_;
    int ox = p - oy * OW_;
    int ky = tap / 3;
    int kx = tap - ky * 3;
    const float4* src =
        (const float4*)&x[(((size_t)(b * H_ + oy + ky) * W_ + ox + kx) * IC_ + i) * IL_];
    float4 v0 = src[0];
    float4 v1 = src[1];
    float4* dst = (float4*)&As[tile][rr][tap * 8];
    dst[0] = v0;
    dst[1] = v1;
  }
  __syncthreads();

  v8f acc[8] = {};

  #pragma unroll
  for (int kk = 0; kk < KTOT / 4; ++kk) {
    const int kb = kk * 4 + half * 2;   // ISA 7.12.2: lanes 0-15 K{0,1}; 16-31 K{2,3}
    v2f afrag;
    afrag.x = As[wave][lm][kb + 0];
    afrag.y = As[wave][lm][kb + 1];
    #pragma unroll
    for (int nt = 0; nt < 8; ++nt) {
      v2f bfrag;
      bfrag.x = Bs[kb + 0][nt * 16 + lm];
      bfrag.y = Bs[kb + 1][nt * 16 + lm];
      acc[nt] = __builtin_amdgcn_wmma_f32_16x16x4_f32(
          false, afrag, false, bfrag, (short)0, acc[nt], false, false);
    }
  }

  // D layout: lanes 0-15: vgpr r -> (M=r, N=lm); lanes 16-31: vgpr r -> (M=8+r, N=lm)
  const int mbase = m0 + wave * 16 + half * 8;
  #pragma unroll
  for (int nt = 0; nt < 8; ++nt) {
    #pragma unroll
    for (int r = 0; r < 8; ++r) {
      uhat[(size_t)(mbase + r) * NCH + nt * 16 + lm] = acc[nt][r];
    }
  }
}

// ------------------------------------------------------------------
// Kernel 2: per (b,p): t[n] = sum_i u_hat ; logits a[b,i,j] = <u_hat, t>/sqrt(8)
// ------------------------------------------------------------------
__global__ __launch_bounds__(256)
void tlogits_kernel(const float* __restrict__ uhat, float* __restrict__ t,
                    float* __restrict__ logits) {
  __shared__ float U[IC_][NCH];   // 16 KB
  __shared__ float T[NCH];
  const int bp  = blockIdx.x;     // b*900 + p
  const int b   = bp / P_;
  const int p   = bp - b * P_;
  const int tid = threadIdx.x;

  for (int idx = tid; idx < IC_ * NCH; idx += 256) {
    int i = idx >> 7;
    int n = idx & 127;
    U[i][n] = uhat[((size_t)(b * IC_ + i) * P_ + p) * NCH + n];
  }
  __syncthreads();

  if (tid < NCH) {
    float s = 0.f;
    #pragma unroll
    for (int i = 0; i < IC_; ++i) s += U[i][tid];
    T[tid] = s;
    t[(size_t)bp * NCH + tid] = s;
  }
  __syncthreads();

  const float inv_sqrt_l = 0.35355339059327379f;   // 1/sqrt(8)
  for (int u = tid; u < IC_ * C_; u += 256) {
    int i = u >> 4;
    int c = u & 15;
    float d = 0.f;
    #pragma unroll
    for (int l = 0; l < L_; ++l) d += U[i][c * 8 + l] * T[c * 8 + l];
    logits[(size_t)(b * IC_ + i) * J_ + p * C_ + c] = d * inv_sqrt_l;
  }
}

// ------------------------------------------------------------------
// Kernel 3: softmax over J=14400 per (b,i) row, in place.
// ------------------------------------------------------------------
__global__ __launch_bounds__(256)
void softmax_kernel(float* __restrict__ a) {
  __shared__ float red[256];
  float* v = a + (size_t)blockIdx.x * J_;
  const int tid = threadIdx.x;

  float m = -INFINITY;
  for (int j = tid; j < J_; j += 256) m = fmaxf(m, v[j]);
  red[tid] = m;
  __syncthreads();
  for (int s = 128; s > 0; s >>= 1) {
    if (tid < s) red[tid] = fmaxf(red[tid], red[tid + s]);
    __syncthreads();
  }
  m = red[0];
  __syncthreads();

  float sum = 0.f;
  for (int j = tid; j < J_; j += 256) sum += __expf(v[j] - m);
  red[tid] = sum;
  __syncthreads();
  for (int s = 128; s > 0; s >>= 1) {
    if (tid < s) red[tid] += red[tid + s];
    __syncthreads();
  }
  const float inv = 1.0f / red[0];

  for (int j = tid; j < J_; j += 256) v[j] = __expf(v[j] - m) * inv;
}

// ------------------------------------------------------------------
// Kernel 4: s[n] = sum_i u_hat[bi,p,n]*a[b,i,p*16+c] + t[n]*b_route[p*128+n];
//           squash over each capsule's 8 lanes; write output.
// ------------------------------------------------------------------
__global__ __launch_bounds__(128)
void route_squash_kernel(const float* __restrict__ uhat, const float* __restrict__ t,
                         const float* __restrict__ a, const float* __restrict__ b_route,
                         float* __restrict__ out) {
  __shared__ float Al[IC_ * C_];   // 512 routing weights for this (b,p)
  __shared__ float S[NCH];
  const int bp  = blockIdx.x;
  const int b   = bp / P_;
  const int p   = bp - b * P_;
  const int tid = threadIdx.x;     // n in [0,128)

  for (int u = tid; u < IC_ * C_; u += 128) {
    int i = u >> 4;
    int c = u & 15;
    Al[u] = a[(size_t)(b * IC_ + i) * J_ + p * C_ + c];
  }
  __syncthreads();

  const int n = tid;
  const int c = n >> 3;
  float s = t[(size_t)bp * NCH + n] * b_route[(size_t)p * NCH + n];
  for (int i = 0; i < IC_; ++i) {
    const float* row = &uhat[((size_t)(b * IC_ + i) * P_ + p) * NCH + n];
    if (i + 1 < IC_)
      __builtin_prefetch(row + (size_t)P_ * NCH, 0, 0);   // global_prefetch_b8: next i row
    s += row[0] * Al[i * C_ + c];
  }
  S[n] = s * s;
  __syncthreads();

  float n2 = 1e-7f;
  #pragma unroll
  for (int l = 0; l < 8; ++l) n2 += S[c * 8 + l];
  const float norm = sqrtf(n2);
  out[(size_t)bp * NCH + n] = (1.0f - __expf(-norm)) * (s / norm);
}

// ------------------------------------------------------------------
extern "C" void kernel_launch(void* const* d_in, const int* in_sizes, int n_in,
                              void* d_out, int out_size, void* d_ws, size_t ws_size,
                              hipStream_t stream) {
  const float* x       = (const float*)d_in[0];
  const float* w       = (const float*)d_in[1];
  const float* b_route = (const float*)d_in[2];
  // d_in[3] = stride (==1, baked into indexing)

  char* ws = (char*)d_ws;
  const size_t uhat_bytes = (size_t)MTOT * NCH * sizeof(float);        // 236 MB
  const size_t t_bytes    = (size_t)B_ * P_ * NCH * sizeof(float);     // 7.4 MB
  float* uhat   = (float*)(ws);
  float* t      = (float*)(ws + uhat_bytes);
  float* logits = (float*)(ws + uhat_bytes + t_bytes);                 // 29.5 MB
  float* out    = (float*)d_out;

  conv_wmma_kernel<<<MTOT / 64, 128, 0, stream>>>(x, w, uhat);
  tlogits_kernel<<<B_ * P_, 256, 0, stream>>>(uhat, t, logits);
  softmax_kernel<<<BI_, 256, 0, stream>>>(logits);
  route_squash_kernel<<<B_ * P_, 128, 0, stream>>>(uhat, t, logits, b_route, out);
}